// DeformableConv3d_63058709840071
// MI455X (gfx1250) — compile-verified
//
#include <hip/hip_runtime.h>

typedef float v2f __attribute__((ext_vector_type(2)));
typedef float v8f __attribute__((ext_vector_type(8)));
typedef float v2fa __attribute__((ext_vector_type(2), aligned(4)));

#define Sdim 48
#define S2   (48 * 48)
#define S3   (48 * 48 * 48)
#define CIN  16
#define OCH  32
#define TAPS 27
#define KDIM 432   // CIN * TAPS
#define OFFCH 81   // 3 * TAPS

// One wave = one strip of 16 consecutive W positions at fixed (n, od, oh).
// Phase 1: offsets via f32 WMMA GEMM (96x432 @ 432x16), staged to LDS.
// Phase 2: trilinear gather (paired-corner b64 loads) + f32 WMMA GEMM
//          (32x432 @ 432x16) -> out.
__global__ __launch_bounds__(128) void deform_conv3d_fused(
    const float* __restrict__ x,       // [2][16][48][48][48]
    const float* __restrict__ weight,  // [32][16][3][3][3]
    const float* __restrict__ offw,    // [81][16][3][3][3]
    const float* __restrict__ offb,    // [81]
    float* __restrict__ out)           // [2][32][48][48][48]
{
    const int lane = threadIdx.x & 31;
    const int wv   = threadIdx.x >> 5;
    const int gid  = blockIdx.x * 4 + wv;

    // strip decode: n, od, oh, owBase
    const int n      = gid / (Sdim * Sdim * 3);
    int rem          = gid - n * (Sdim * Sdim * 3);
    const int od     = rem / (Sdim * 3);
    rem              = rem - od * (Sdim * 3);
    const int oh     = rem / 3;
    const int owBase = (rem - oh * 3) * 16;

    const int nn = lane & 15;   // N-column of the WMMA tile = spatial position
    const int hi = lane >> 4;   // half-wave selector (K pairs)

    const float* __restrict__ xn = x + n * CIN * S3;

    // weight tensors are tiny and reused by every block: pull toward L2/L1
    __builtin_prefetch(weight, 0, 1);
    __builtin_prefetch(offw, 0, 1);

    __shared__ float off_lds[4][96][16];

    // ---------------- Phase 1: offset GEMM ----------------
    // D[96x16] = offw[96x432(padded)] * im2col(x)[432x16], K order: c*27 + tap
    v8f acc0 = {0.f,0.f,0.f,0.f,0.f,0.f,0.f,0.f};
    v8f acc[6];
    #pragma unroll
    for (int i = 0; i < 6; ++i) acc[i] = acc0;

    const int odm1 = od - 1;
    const int ohm1 = oh - 1;
    const int owm1 = owBase + nn - 1;

    // incremental decomposition of k = 4*kq + hi*2 into (c, t): avoids /27
    int tL = hi * 2;   // t for j=0 (k < 27 initially -> c = 0)
    int cL = 0;

    #pragma unroll 1
    for (int kq = 0; kq < KDIM / 4; ++kq) {
        const int kbase = kq * 4 + hi * 2;

        // B fragment: branchless zero-padded im2col gathers (clamp + select)
        v2f b;
        {
            const int t  = tL;
            const int kd = t / 9;
            const int r9 = t - kd * 9;
            const int kh = r9 / 3;
            const int kw = r9 - kh * 3;
            const int zd = odm1 + kd;
            const int zh = ohm1 + kh;
            const int zw = owm1 + kw;
            const bool ok = ((unsigned)zd < 48u) & ((unsigned)zh < 48u) &
                            ((unsigned)zw < 48u);
            const int ad = min(max(zd, 0), 47);
            const int ah = min(max(zh, 0), 47);
            const int aw = min(max(zw, 0), 47);
            const float v = xn[cL * S3 + ad * S2 + ah * Sdim + aw];
            b[0] = ok ? v : 0.f;
        }
        {
            const int wrap = (tL == 26);
            const int t  = wrap ? 0 : tL + 1;
            const int c  = cL + wrap;
            const int kd = t / 9;
            const int r9 = t - kd * 9;
            const int kh = r9 / 3;
            const int kw = r9 - kh * 3;
            const int zd = odm1 + kd;
            const int zh = ohm1 + kh;
            const int zw = owm1 + kw;
            const bool ok = ((unsigned)zd < 48u) & ((unsigned)zh < 48u) &
                            ((unsigned)zw < 48u);
            const int ad = min(max(zd, 0), 47);
            const int ah = min(max(zh, 0), 47);
            const int aw = min(max(zw, 0), 47);
            const float v = xn[c * S3 + ad * S2 + ah * Sdim + aw];
            b[1] = ok ? v : 0.f;
        }

        // A fragments (b64 loads, K contiguous in natural layout) + WMMA
        #pragma unroll
        for (int mt = 0; mt < 6; ++mt) {
            const int m  = mt * 16 + nn;
            const int mc = m > 80 ? 80 : m;            // clamp padded rows
            const float* ap = offw + mc * KDIM + kbase;
            v2f a;
            a[0] = ap[0];
            a[1] = ap[1];
            if (m > 80) { a[0] = 0.f; a[1] = 0.f; }    // zero the pad rows
            acc[mt] = __builtin_amdgcn_wmma_f32_16x16x4_f32(
                false, a, false, b, (short)0, acc[mt], false, false);
        }

        // advance k by 4: (c, t) wrap
        tL += 4;
        if (tL >= TAPS) { tL -= TAPS; cL += 1; }
    }

    // Transpose accumulators (C layout: vgpr=v -> M=v+8*hi, lane&15 -> N) to LDS
    #pragma unroll
    for (int mt = 0; mt < 6; ++mt) {
        #pragma unroll
        for (int v = 0; v < 8; ++v) {
            const int m = mt * 16 + v + hi * 8;
            const float bia = (m < OFFCH) ? offb[m] : 0.f;
            off_lds[wv][m][nn] = acc[mt][v] + bia;
        }
    }
    asm volatile("s_wait_dscnt 0" ::: "memory");   // wave-local LDS RAW fence

    // ---------------- Phase 2: trilinear sample + output GEMM ----------------
    // D[32x16] = W[32x432] * sampled[432x16], K order: tap*16 + c
    v8f oacc[2];
    oacc[0] = acc0;
    oacc[1] = acc0;

    #pragma unroll 1
    for (int t = 0; t < TAPS; ++t) {
        const int kd = t / 9;
        const int r9 = t - kd * 9;
        const int kh = r9 / 3;
        const int kw = r9 - kh * 3;

        const float offD = off_lds[wv][t][nn];
        const float offH = off_lds[wv][TAPS + t][nn];
        const float offW = off_lds[wv][2 * TAPS + t][nn];

        const float pd = (float)(odm1 + kd) + offD;
        const float ph = (float)(ohm1 + kh) + offH;
        const float pw = (float)(owm1 + kw) + offW;

        const float dff = floorf(pd);
        const float hff = floorf(ph);
        const float wff = floorf(pw);
        const float fd = pd - dff;
        const float fh = ph - hff;
        const float fw = pw - wff;
        const int d0 = (int)dff;
        const int h0 = (int)hff;
        const int w0 = (int)wff;

        // paired-corner handling along W: both ww corners live at
        // [wlo, wlo+1]; boundary cases folded into the weights.
        const int  wlo     = min(max(w0, 0), Sdim - 2);
        const bool selNorm = (w0 == wlo);        // interior
        const bool selLow  = (w0 == wlo - 1);    // w0 == -1
        const bool selHigh = (w0 == wlo + 1);    // w0 == 47
        const bool okW0    = ((unsigned)w0 < 48u);
        const bool okW1    = ((unsigned)(w0 + 1) < 48u);

        float wA[4], wB[4];
        int   cof[4];
        #pragma unroll
        for (int dh = 0; dh < 4; ++dh) {
            const int dd = dh >> 1;
            const int hh = dh & 1;
            const int di = d0 + dd;
            const int hy = h0 + hh;
            const bool okdh = ((unsigned)di < 48u) & ((unsigned)hy < 48u);
            const float wdh = (dd ? fd : 1.f - fd) * (hh ? fh : 1.f - fh);
            const float g0 = (okdh & okW0) ? wdh * (1.f - fw) : 0.f;
            const float g1 = (okdh & okW1) ? wdh * fw         : 0.f;
            wA[dh] = selNorm ? g0 : (selLow  ? g1 : 0.f);  // weight on elem wlo
            wB[dh] = selNorm ? g1 : (selHigh ? g0 : 0.f);  // weight on elem wlo+1
            const int cd = min(max(di, 0), 47);
            const int ch = min(max(hy, 0), 47);
            cof[dh] = cd * S2 + ch * Sdim + wlo;
        }

        #pragma unroll
        for (int cq = 0; cq < 4; ++cq) {
            const int c0 = cq * 4 + hi * 2;

            // B fragment: trilinear-sampled values for 2 channels,
            // 4 paired (2-wide) gathers per channel instead of 8 scalar ones
            v2f b;
            #pragma unroll
            for (int j = 0; j < 2; ++j) {
                const float* __restrict__ xc = xn + (c0 + j) * S3;
                float s = 0.f;
                #pragma unroll
                for (int dh = 0; dh < 4; ++dh) {
                    const v2fa e = *(const v2fa*)(xc + cof[dh]);
                    s = fmaf(wA[dh], e[0], s);
                    s = fmaf(wB[dh], e[1], s);
                }
                b[j] = s;
            }

            // A fragments from weight[o][c][t] (flat o*432 + c*27 + t)
            #pragma unroll
            for (int mt = 0; mt < 2; ++mt) {
                const int m = mt * 16 + nn;
                const float* wp = weight + m * KDIM + t;
                v2f a;
                a[0] = wp[c0 * TAPS];
                a[1] = wp[(c0 + 1) * TAPS];
                oacc[mt] = __builtin_amdgcn_wmma_f32_16x16x4_f32(
                    false, a, false, b, (short)0, oacc[mt], false, false);
            }
        }
    }

    // ---------------- Write out ----------------
    float* __restrict__ op = out + (n * OCH) * S3 + od * S2 + oh * Sdim + owBase + nn;
    #pragma unroll
    for (int mt = 0; mt < 2; ++mt) {
        #pragma unroll
        for (int v = 0; v < 8; ++v) {
            const int o = mt * 16 + v + hi * 8;
            op[o * S3] = oacc[mt][v];
        }
    }
}

extern "C" void kernel_launch(void* const* d_in, const int* in_sizes, int n_in,
                              void* d_out, int out_size, void* d_ws, size_t ws_size,
                              hipStream_t stream) {
    const float* x    = (const float*)d_in[0];  // (2,16,48,48,48)
    const float* w    = (const float*)d_in[1];  // (32,16,3,3,3)
    const float* offw = (const float*)d_in[2];  // (81,16,3,3,3)
    const float* offb = (const float*)d_in[3];  // (81,)
    float* out = (float*)d_out;                 // (2,32,48,48,48)

    const int strips = 2 * Sdim * Sdim * 3;     // 13824 strips, 4 waves/block
    dim3 grid(strips / 4);
    dim3 block(128);
    hipLaunchKernelGGL(deform_conv3d_fused, grid, block, 0, stream,
                       x, w, offw, offb, out);
}